// Q_Model_3Layer_89816356094170
// MI455X (gfx1250) — compile-verified
//
#include <hip/hip_runtime.h>
#include <math.h>

// ---------------------------------------------------------------------------
// out(x) = 0.5*(<Z>(x)+1) is exactly a degree-3 trig polynomial in x:
//   out(x) = w0 + w1 cos x + w2 sin x + w3 cos 2x + w4 sin 2x + w5 cos 3x + w6 sin 3x
// Coefficients depend only on params; recovered exactly via 8-point DFT of the
// reference circuit evaluated at x_j = 2*pi*j/8 (fp64, single thread).
// ---------------------------------------------------------------------------

#define TWO_PI_D 6.283185307179586476925286766559

__global__ void setup_coeffs(const float* __restrict__ params,
                             float* __restrict__ w)
{
    if (threadIdx.x != 0 || blockIdx.x != 0) return;

    double z[8];
    for (int j = 0; j < 8; ++j) {
        const double x  = (TWO_PI_D / 8.0) * (double)j;
        const double ch = cos(0.5 * x), sh = sin(0.5 * x);
        // state (a, b) = (1, 0)
        double ar = 1.0, ai = 0.0, br = 0.0, bi = 0.0;
        for (int l = 0; l < 3; ++l) {
            // encoder RY(x)
            double nar = ch * ar - sh * br, nai = ch * ai - sh * bi;
            double nbr = sh * ar + ch * br, nbi = sh * ai + ch * bi;
            ar = nar; ai = nai; br = nbr; bi = nbi;
            // RZ(p0): a *= e^{-ih}, b *= e^{+ih}
            double h  = 0.5 * (double)params[3 * l + 0];
            double cp = cos(h), sp = sin(h);
            nar = cp * ar + sp * ai;  nai = cp * ai - sp * ar;
            nbr = cp * br - sp * bi;  nbi = cp * bi + sp * br;
            ar = nar; ai = nai; br = nbr; bi = nbi;
            // RY(p1)
            h = 0.5 * (double)params[3 * l + 1];
            double cy = cos(h), sy = sin(h);
            nar = cy * ar - sy * br; nai = cy * ai - sy * bi;
            nbr = sy * ar + cy * br; nbi = sy * ai + cy * bi;
            ar = nar; ai = nai; br = nbr; bi = nbi;
            // RZ(p2)
            h  = 0.5 * (double)params[3 * l + 2];
            cp = cos(h); sp = sin(h);
            nar = cp * ar + sp * ai;  nai = cp * ai - sp * ar;
            nbr = cp * br - sp * bi;  nbi = cp * bi + sp * br;
            ar = nar; ai = nai; br = nbr; bi = nbi;
        }
        const double zz = (ar * ar + ai * ai) - (br * br + bi * bi);
        z[j] = 0.5 * (zz + 1.0);
    }

    // 8-point real DFT -> harmonics 0..3 (exact for band limit 3)
    double c[7];
    double A0 = 0.0;
    for (int j = 0; j < 8; ++j) A0 += z[j];
    c[0] = A0 * 0.125;
    for (int n = 1; n <= 3; ++n) {
        double An = 0.0, Bn = 0.0;
        for (int j = 0; j < 8; ++j) {
            const double ang = (TWO_PI_D / 8.0) * (double)(n * j);
            An += z[j] * cos(ang);
            Bn += z[j] * sin(ang);
        }
        c[2 * n - 1] = An * 0.25;
        c[2 * n]     = Bn * 0.25;
    }
    for (int k = 0; k < 7; ++k) w[k] = (float)c[k];
}

// ---------------------------------------------------------------------------
// Streaming evaluation: 2 transcendentals + 11 FMAs per element.
// ---------------------------------------------------------------------------

__device__ __forceinline__ float eval_elem(float xi,
    float w0, float w1, float w2, float w3, float w4, float w5, float w6)
{
    const float s1 = __sinf(xi);
    const float c1 = __cosf(xi);
    const float c2 = __builtin_fmaf(2.0f * c1, c1, -1.0f);   // cos 2x
    const float s2 = 2.0f * s1 * c1;                         // sin 2x
    const float c3 = __builtin_fmaf(c2, c1, -(s2 * s1));     // cos 3x
    const float s3 = __builtin_fmaf(s2, c1,  (c2 * s1));     // sin 3x
    float acc = __builtin_fmaf(w1, c1, w0);
    acc = __builtin_fmaf(w2, s1, acc);
    acc = __builtin_fmaf(w3, c2, acc);
    acc = __builtin_fmaf(w4, s2, acc);
    acc = __builtin_fmaf(w5, c3, acc);
    acc = __builtin_fmaf(w6, s3, acc);
    return acc;
}

__global__ __launch_bounds__(256) void qreupload_eval(
    const float* __restrict__ x, const float* __restrict__ w,
    float* __restrict__ out, int n)
{
    __shared__ float tile[1024];                 // 256 threads * 16B
    const int tid = threadIdx.x;
    const long long base = (long long)blockIdx.x * 1024LL + (long long)tid * 4LL;

    // 7 uniform coefficients (scalar-loadable, L2-resident)
    const float w0 = w[0], w1 = w[1], w2 = w[2], w3 = w[3],
                w4 = w[4], w5 = w[5], w6 = w[6];

    if (base + 4LL <= (long long)n) {
        // CDNA5 async global->LDS copy (ASYNCcnt path), 16 bytes per lane.
        const unsigned lds_off = (unsigned)(size_t)(&tile[tid * 4]);
        const float*   gp      = x + base;
        asm volatile("global_load_async_to_lds_b128 %0, %1, off"
                     :
                     : "v"(lds_off), "v"(gp)
                     : "memory");
        asm volatile("s_wait_asynccnt 0x0" ::: "memory");
        // each lane reads back exactly the 16B it staged -> no barrier needed
        const float4 v = *reinterpret_cast<const float4*>(&tile[tid * 4]);

        float4 r;
        r.x = eval_elem(v.x, w0, w1, w2, w3, w4, w5, w6);
        r.y = eval_elem(v.y, w0, w1, w2, w3, w4, w5, w6);
        r.z = eval_elem(v.z, w0, w1, w2, w3, w4, w5, w6);
        r.w = eval_elem(v.w, w0, w1, w2, w3, w4, w5, w6);
        *reinterpret_cast<float4*>(out + base) = r;
    } else {
        // tail fallback
        for (int k = 0; k < 4; ++k) {
            const long long i = base + k;
            if (i < (long long)n)
                out[i] = eval_elem(x[i], w0, w1, w2, w3, w4, w5, w6);
        }
    }
}

// ---------------------------------------------------------------------------

extern "C" void kernel_launch(void* const* d_in, const int* in_sizes, int n_in,
                              void* d_out, int out_size, void* d_ws, size_t ws_size,
                              hipStream_t stream)
{
    const float* x      = (const float*)d_in[0];   // [B] encoding angles
    const float* params = (const float*)d_in[1];   // [3,3] gate angles
    float*       w      = (float*)d_ws;            // 7 Fourier coefficients
    const int    n      = in_sizes[0];
    if (n <= 0) return;

    setup_coeffs<<<1, 1, 0, stream>>>(params, w);

    const int blocks = (n + 1023) / 1024;          // 1024 elements per block
    qreupload_eval<<<blocks, 256, 0, stream>>>(x, w, (float*)d_out, n);
}